// PLRNN_88459146428768
// MI455X (gfx1250) — compile-verified
//
#include <hip/hip_runtime.h>

#define DZ 64
#define DXX 128
#define T_STEPS 200000

typedef __attribute__((ext_vector_type(2))) float v2f;
typedef __attribute__((ext_vector_type(8))) float v8f;

// ---------------------------------------------------------------------------
// Kernel 1: sequential PLRNN scan. One workgroup, 128 threads = 4 waves
// (one wave per SIMD32 of the WGP). thread = (row 0..63, half 0..1).
// Each thread keeps 32 W coefficients in VGPRs; relu(z) is broadcast through
// a double-buffered LDS array with ONE barrier per step. Writes Z in (dz,T)
// layout (stride-T per row) which stays resident in the 192MB L2 for kernel 2.
// ---------------------------------------------------------------------------
__global__ __launch_bounds__(128) void plrnn_scan_kernel(
    const float* __restrict__ AW, const float* __restrict__ h,
    const float* __restrict__ Q0, const float* __restrict__ Q,
    const float* __restrict__ eps_z, const float* __restrict__ z0_eps,
    float* __restrict__ Zout) {
  const int tid  = threadIdx.x;
  const int row  = tid >> 1;        // 0..63
  const int part = tid & 1;         // 0..1 (column half)
  const int cbase = part * 32;

  // Register-resident slice of W (row `row`, columns cbase..cbase+31), diag zeroed.
  float w[32];
#pragma unroll
  for (int j = 0; j < 32; ++j) {
    const int col = cbase + j;
    const float v = AW[row * DZ + col];
    w[j] = (col == row) ? 0.0f : v;
  }
  const float dA = AW[row * DZ + row];
  const float hr = h[row];
  const float Qr = Q[row];
  const float* __restrict__ epsRow = eps_z + (size_t)row * T_STEPS;
  float* __restrict__ zRow = Zout + (size_t)row * T_STEPS;

  __shared__ __align__(16) float zr[2][DZ];   // relu(z), double-buffered

  float zcur = 0.0f;
  if (part == 0) {
    zcur = Q0[row] * z0_eps[row];
    zr[0][row] = fmaxf(zcur, 0.0f);
  }
  __syncthreads();

  int buf = 0;
  for (int t = 0; t < T_STEPS; ++t) {
    // prefetch next eps cacheline once per 16 steps (global_prefetch_b8)
    if (part == 0 && (t & 15) == 0) {
      __builtin_prefetch(epsRow + t + 16, 0, 0);
    }
    const float* __restrict__ zb = &zr[buf][cbase];
    // 32-deep dot product, 4 accumulators to pipeline FMA latency
    float a0 = 0.f, a1 = 0.f, a2 = 0.f, a3 = 0.f;
#pragma unroll
    for (int j = 0; j < 32; j += 4) {
      a0 = fmaf(w[j + 0], zb[j + 0], a0);
      a1 = fmaf(w[j + 1], zb[j + 1], a1);
      a2 = fmaf(w[j + 2], zb[j + 2], a2);
      a3 = fmaf(w[j + 3], zb[j + 3], a3);
    }
    float acc = (a0 + a1) + (a2 + a3);
    acc += __shfl_xor(acc, 1);      // combine the two column halves (wave32)

    if (part == 0) {
      const float zn = fmaf(dA, zcur, acc) + hr + Qr * epsRow[t];
      zcur = zn;
      zr[buf ^ 1][row] = fmaxf(zn, 0.0f);
      zRow[t] = zn;                 // Z.T output, stays hot in L2
    }
    __syncthreads();                // single barrier/step (double buffer)
    buf ^= 1;
  }
}

// ---------------------------------------------------------------------------
// Kernel 2: emission GEMM  X.T (128 x T) = B (128x64) @ Z (64xT) + R*eta,
// using V_WMMA_F32_16X16X4_F32 (full fp32 precision).
// Block = 256 threads = 8 waves; wave w owns x-tile rows [16w,16w+16).
// Each wave hoists all 16 A-fragments of B (32 VGPRs) and sweeps 4 t-tiles.
// M = x (dx rows), N = t (time), K = z. 16 WMMAs per output tile (K=64/4).
// ---------------------------------------------------------------------------
__global__ __launch_bounds__(256) void plrnn_emit_kernel(
    const float* __restrict__ Bm, const float* __restrict__ R,
    const float* __restrict__ eta, const float* __restrict__ Zbuf,
    float* __restrict__ Xout) {
  const int lane = threadIdx.x & 31;
  const int wave = threadIdx.x >> 5;   // 0..7 -> x-tile
  const int x0   = wave * 16;
  const int n    = lane & 15;          // N index within tile
  const int hi   = lane >> 4;          // lane-half selects K pairs / M+8 rows

  // A fragments: 16x4 tiles of B for K = 4*kk .. 4*kk+3.
  // lane<16: K = 4kk+{0,1}; lane>=16: K = 4kk+{2,3}   (ISA 7.12.2, 32-bit A 16x4)
  v2f a[16];
  {
    const int m = x0 + n;
#pragma unroll
    for (int kk = 0; kk < 16; ++kk) {
      const float* p = Bm + m * DZ + kk * 4 + hi * 2;   // 8B aligned
      a[kk] = *reinterpret_cast<const v2f*>(p);
    }
  }

  const long t0base = (long)blockIdx.x * 64;            // 4 t-tiles per block
  for (int tt = 0; tt < 4; ++tt) {
    const long t0 = t0base + tt * 16;
    v8f c = {0.f, 0.f, 0.f, 0.f, 0.f, 0.f, 0.f, 0.f};
#pragma unroll
    for (int kk = 0; kk < 16; ++kk) {
      // B fragment: 4x16 tile of Z. VGPR0 holds rows K,K+2; VGPR1 rows K+1,K+3
      const int krow = kk * 4 + hi * 2;
      v2f b;
      b.x = Zbuf[(size_t)krow * T_STEPS + t0 + n];
      b.y = Zbuf[(size_t)(krow + 1) * T_STEPS + t0 + n];
      // (neg_a, A, neg_b, B, c_mod, C, reuse_a, reuse_b)
      c = __builtin_amdgcn_wmma_f32_16x16x4_f32(
          false, a[kk], false, b, (short)0, c, false, false);
    }
    // D layout: vgpr r -> x-row x0+r (lanes 0-15) / x0+r+8 (lanes 16-31)
#pragma unroll
    for (int r = 0; r < 8; ++r) {
      const int xr = x0 + r + hi * 8;
      const size_t idx = (size_t)xr * T_STEPS + (size_t)(t0 + n);
      Xout[idx] = c[r] + R[xr] * eta[idx];
    }
  }
}

// ---------------------------------------------------------------------------
extern "C" void kernel_launch(void* const* d_in, const int* in_sizes, int n_in,
                              void* d_out, int out_size, void* d_ws, size_t ws_size,
                              hipStream_t stream) {
  const float* AW     = (const float*)d_in[0];   // (64,64)
  const float* h      = (const float*)d_in[1];   // (64,)
  const float* Bm     = (const float*)d_in[2];   // (128,64)
  const float* Q0     = (const float*)d_in[3];   // (64,)
  const float* Q      = (const float*)d_in[4];   // (64,)
  const float* R      = (const float*)d_in[5];   // (128,)
  const float* eps_z  = (const float*)d_in[6];   // (64,T)
  const float* eta_x  = (const float*)d_in[7];   // (128,T)
  const float* z0_eps = (const float*)d_in[8];   // (64,)

  float* Xout = (float*)d_out;                        // X.T : (128, T)
  float* Zout = Xout + (size_t)DXX * T_STEPS;         // Z.T : (64, T)

  // 1) sequential scan -> Z (latency-bound; one WGP, 4 waves)
  plrnn_scan_kernel<<<1, 128, 0, stream>>>(AW, h, Q0, Q, eps_z, z0_eps, Zout);

  // 2) emission GEMM with fp32 WMMA (bandwidth-bound; Z read from L2)
  const int tTiles = T_STEPS / 16;                    // 12500
  plrnn_emit_kernel<<<tTiles / 4, 256, 0, stream>>>(Bm, R, eta_x, Zout, Xout);
}